// CrossFrameAttention_88691074662881
// MI455X (gfx1250) — compile-verified
//
#include <hip/hip_runtime.h>
#include <math.h>

// ---- sizes ----
#define B_  16
#define S_  6
#define E_  256
#define HW_ 4096              // 64*64 spatial
#define NH_ 8
#define HD_ 32
#define ROWS_ (B_*S_)         // 96
#define CH_   (ROWS_*E_)      // 24576 channels (b,s,e)
#define CROSS_ELEMS_ ((size_t)CH_ * HW_)   // 100,663,296

typedef float flt2 __attribute__((ext_vector_type(2)));
typedef float flt4 __attribute__((ext_vector_type(4)));
typedef float flt8 __attribute__((ext_vector_type(8)));

// ---------------------------------------------------------------------------
// K1: g[b,s,e] = mean_{h,w} ff + temporal_pos[s,e]
// One wave per channel; 32 lanes x 32 float4 NT loads, shuffle reduction.
// ---------------------------------------------------------------------------
__global__ void cfa_mean_kernel(const float* __restrict__ ff,
                                const float* __restrict__ tpos,
                                float* __restrict__ g) {
  const int wave = (blockIdx.x * blockDim.x + threadIdx.x) >> 5;   // 0..24575
  const int lane = threadIdx.x & 31;
  const flt4* p = (const flt4*)(ff + (size_t)wave * HW_);
  float sum = 0.f;
#pragma unroll
  for (int i = 0; i < 32; ++i) {
    flt4 x = __builtin_nontemporal_load(p + i * 32 + lane);
    sum += (x.x + x.y) + (x.z + x.w);
  }
  sum += __shfl_xor(sum, 16, 32);
  sum += __shfl_xor(sum,  8, 32);
  sum += __shfl_xor(sum,  4, 32);
  sum += __shfl_xor(sum,  2, 32);
  sum += __shfl_xor(sum,  1, 32);
  if (lane == 0) {
    int e = wave & (E_ - 1);
    int s = (wave >> 8) % S_;
    g[wave] = sum * (1.0f / (float)HW_) + tpos[s * E_ + e];
  }
}

// ---------------------------------------------------------------------------
// WMMA helper: one wave computes a 16x16 tile of  OUT = A(96x256) @ W^T + bias
// using V_WMMA_F32_16X16X4_F32, K=256 in 64 steps.
// ---------------------------------------------------------------------------
__device__ __forceinline__ void gemm_tile_f32(const float* __restrict__ A,
                                              const float* __restrict__ W,
                                              const float* __restrict__ bias,
                                              float* __restrict__ OUT,
                                              int tm, int tn, int lane) {
  const int l15 = lane & 15;
  const int kk  = (lane >> 4) << 1;           // 0 or 2
  const float* arow = A + (size_t)(tm * 16 + l15) * E_ + kk;  // A[row][k]
  const float* brow = W + (size_t)(tn * 16 + l15) * E_ + kk;  // B[k][n] = W[n][k]
  const float bval = bias[tn * 16 + l15];
  flt8 acc = {bval, bval, bval, bval, bval, bval, bval, bval};
#pragma unroll 8
  for (int kb = 0; kb < E_; kb += 4) {
    flt2 a; a.x = arow[kb]; a.y = arow[kb + 1];
    flt2 b; b.x = brow[kb]; b.y = brow[kb + 1];
    acc = __builtin_amdgcn_wmma_f32_16x16x4_f32(false, a, false, b,
                                                (short)0, acc, false, false);
  }
  const int rowoff = (lane >> 4) * 8;
#pragma unroll
  for (int r = 0; r < 8; ++r)
    OUT[(size_t)(tm * 16 + rowoff + r) * E_ + tn * 16 + l15] = acc[r];
}

// K2: q,k,v projections. 3 proj * 6 * 16 tiles = 288 waves (36 blocks x 8 waves)
__global__ void cfa_qkv_kernel(const float* __restrict__ g,
                               const float* __restrict__ Wq, const float* __restrict__ bq,
                               const float* __restrict__ Wk, const float* __restrict__ bk,
                               const float* __restrict__ Wv, const float* __restrict__ bv,
                               float* __restrict__ q, float* __restrict__ k,
                               float* __restrict__ v) {
  const int wave = (blockIdx.x * blockDim.x + threadIdx.x) >> 5;  // 0..287
  const int lane = threadIdx.x & 31;
  const int proj = wave / 96;
  const int t    = wave % 96;
  const float* W    = (proj == 0) ? Wq : (proj == 1) ? Wk : Wv;
  const float* bias = (proj == 0) ? bq : (proj == 1) ? bk : bv;
  float*       OUT  = (proj == 0) ? q  : (proj == 1) ? k  : v;
  gemm_tile_f32(g, W, bias, OUT, t >> 4, t & 15, lane);
}

// K4: out_g = attended @ Wo^T + bo. 96 waves (12 blocks x 8 waves)
__global__ void cfa_oproj_kernel(const float* __restrict__ att,
                                 const float* __restrict__ Wo,
                                 const float* __restrict__ bo,
                                 float* __restrict__ out_g) {
  const int wave = (blockIdx.x * blockDim.x + threadIdx.x) >> 5;  // 0..95
  const int lane = threadIdx.x & 31;
  gemm_tile_f32(att, Wo, bo, out_g, wave >> 4, wave & 15, lane);
}

// ---------------------------------------------------------------------------
// K3: per-(b,h) causal softmax attention. One wave per (b,h): lane = head dim.
// scores via full-wave shuffle reduction; softmax replicated on all lanes.
// ---------------------------------------------------------------------------
__global__ void cfa_attn_kernel(const float* __restrict__ q,
                                const float* __restrict__ k,
                                const float* __restrict__ v,
                                float* __restrict__ attended,
                                float* __restrict__ attn_out) {
  const int wave = (blockIdx.x * blockDim.x + threadIdx.x) >> 5;  // 0..127
  const int lane = threadIdx.x & 31;
  const int b = wave >> 3, h = wave & 7;
  const int col = h * HD_ + lane;

  float qr[S_], kr[S_], vr[S_];
#pragma unroll
  for (int s = 0; s < S_; ++s) {
    const int r = (b * S_ + s) * E_ + col;
    qr[s] = q[r]; kr[s] = k[r]; vr[s] = v[r];
  }

  const float inv_scale = 0.17677669529663687f;   // 1/sqrt(32)
  float p[S_][S_];
#pragma unroll
  for (int s = 0; s < S_; ++s) {
#pragma unroll
    for (int t = 0; t < S_; ++t) {
      float x = qr[s] * kr[t];
      x += __shfl_xor(x, 16, 32);
      x += __shfl_xor(x,  8, 32);
      x += __shfl_xor(x,  4, 32);
      x += __shfl_xor(x,  2, 32);
      x += __shfl_xor(x,  1, 32);
      p[s][t] = x * inv_scale;
    }
  }
#pragma unroll
  for (int s = 0; s < S_; ++s) {
    float m = p[s][0];
#pragma unroll
    for (int t = 1; t < S_; ++t) if (t <= s) m = fmaxf(m, p[s][t]);
    float d = 0.f;
#pragma unroll
    for (int t = 0; t < S_; ++t) {
      float e = (t <= s) ? __expf(p[s][t] - m) : 0.f;
      p[s][t] = e; d += e;
    }
    const float inv = 1.0f / d;
#pragma unroll
    for (int t = 0; t < S_; ++t) p[s][t] *= inv;
  }
  // attended[s][d] = sum_{t<=s} p[s][t] * v[t][d]
#pragma unroll
  for (int s = 0; s < S_; ++s) {
    float a = 0.f;
#pragma unroll
    for (int t = 0; t < S_; ++t) if (t <= s) a += p[s][t] * vr[t];
    attended[(b * S_ + s) * E_ + col] = a;
  }
  // attn probs -> d_out tail (identical on all lanes; lane 0 writes)
  if (lane == 0) {
#pragma unroll
    for (int s = 0; s < S_; ++s)
#pragma unroll
      for (int t = 0; t < S_; ++t)
        attn_out[((b * NH_ + h) * S_ + s) * S_ + t] = p[s][t];
  }
}

// ---------------------------------------------------------------------------
// K5: cross = ff + out_g broadcast over HxW. Grid-stride float4, NT hints.
// ---------------------------------------------------------------------------
__global__ void cfa_cross_kernel(const float* __restrict__ ff,
                                 const float* __restrict__ out_g,
                                 float* __restrict__ out) {
  const size_t n4 = CROSS_ELEMS_ / 4;          // 1024 float4 per channel
  size_t i = (size_t)blockIdx.x * blockDim.x + threadIdx.x;
  const size_t stride = (size_t)gridDim.x * blockDim.x;
  const flt4* in4 = (const flt4*)ff;
  flt4* o4 = (flt4*)out;
  for (; i < n4; i += stride) {
    const float gadd = out_g[i >> 10];
    flt4 x = __builtin_nontemporal_load(in4 + i);
    x.x += gadd; x.y += gadd; x.z += gadd; x.w += gadd;
    __builtin_nontemporal_store(x, o4 + i);
  }
}

// ---------------------------------------------------------------------------
extern "C" void kernel_launch(void* const* d_in, const int* in_sizes, int n_in,
                              void* d_out, int out_size, void* d_ws, size_t ws_size,
                              hipStream_t stream) {
  const float* ff   = (const float*)d_in[0];
  const float* Wq   = (const float*)d_in[1];
  const float* bq   = (const float*)d_in[2];
  const float* Wk   = (const float*)d_in[3];
  const float* bk   = (const float*)d_in[4];
  const float* Wv   = (const float*)d_in[5];
  const float* bv   = (const float*)d_in[6];
  const float* Wo   = (const float*)d_in[7];
  const float* bo   = (const float*)d_in[8];
  const float* tpos = (const float*)d_in[9];

  float* ws    = (float*)d_ws;        // 6 * 24576 floats = 576 KB scratch
  float* g     = ws;
  float* q     = ws + 1 * CH_;
  float* k     = ws + 2 * CH_;
  float* v     = ws + 3 * CH_;
  float* att   = ws + 4 * CH_;
  float* out_g = ws + 5 * CH_;

  float* cross    = (float*)d_out;
  float* attn_out = cross + CROSS_ELEMS_;

  // 24576 waves: one per (b,s,e) channel
  cfa_mean_kernel<<<CH_ / 8, 256, 0, stream>>>(ff, tpos, g);
  // 288 WMMA waves (3 projections x 96 tiles)
  cfa_qkv_kernel<<<36, 256, 0, stream>>>(g, Wq, bq, Wk, bk, Wv, bv, q, k, v);
  // 128 waves: one per (b,h)
  cfa_attn_kernel<<<16, 256, 0, stream>>>(q, k, v, att, attn_out);
  // 96 WMMA waves
  cfa_oproj_kernel<<<12, 256, 0, stream>>>(att, Wo, bo, out_g);
  // streaming broadcast add over 402 MB
  cfa_cross_kernel<<<8192, 256, 0, stream>>>(ff, out_g, cross);
}